// SimpleGNN_8598524526604
// MI455X (gfx1250) — compile-verified
//
#include <hip/hip_runtime.h>
#include <hip/hip_bf16.h>

typedef __attribute__((ext_vector_type(2))) float v2f;
typedef __attribute__((ext_vector_type(8))) float v8f;

__device__ __forceinline__ void atomAddF(float* p, float v) {
    __hip_atomic_fetch_add(p, v, __ATOMIC_RELAXED, __HIP_MEMORY_SCOPE_AGENT);
}

// ---------------- degree / normalization ----------------

__global__ void k_init_deg(float* deg, int n) {
    int i = blockIdx.x * blockDim.x + threadIdx.x;
    if (i < n) deg[i] = 1.0f;   // self-loop
}

__global__ void k_count_deg(const long long* __restrict__ dst, float* deg, int e) {
    int i = blockIdx.x * blockDim.x + threadIdx.x;
    if (i < e) atomAddF(&deg[(int)dst[i]], 1.0f);
}

__global__ void k_finalize_dinv(float* deg, int n) {
    int i = blockIdx.x * blockDim.x + threadIdx.x;
    if (i < n) deg[i] = __frsqrt_rn(deg[i]);   // deg >= 1 always
}

// ---------------- operand prep: pad/transpose so WMMA loops are branch-free ----------------

// xp[N][12]: x with K padded 10 -> 12 (zeros)
__global__ void k_pad_x(const float* __restrict__ x, float* __restrict__ xp, int n) {
    int idx = blockIdx.x * blockDim.x + threadIdx.x;   // over n*12
    if (idx >= n * 12) return;
    int i = idx / 12, k = idx % 12;
    xp[idx] = (k < 10) ? x[i * 10 + k] : 0.f;
}

// Wc1[16][12]: W1 transposed (col-major) + K padded; Wc2[16][16]: W2 transposed, N padded 2 -> 16
__global__ void k_prep_w(const float* __restrict__ W1, const float* __restrict__ W2,
                         float* __restrict__ Wc1, float* __restrict__ Wc2) {
    int idx = threadIdx.x;                 // one block of 256
    if (idx < 192) {                       // 16 x 12
        int m = idx / 12, k = idx % 12;
        Wc1[idx] = (k < 10) ? W1[k * 16 + m] : 0.f;
    }
    // 16 x 16
    {
        int m = idx / 16, k = idx % 16;
        Wc2[idx] = (m < 2) ? W2[k * 2 + m] : 0.f;
    }
}

// ---------------- layer 1 GEMM: h1 = x @ W1 via V_WMMA_F32_16X16X4_F32 ----------------
// One wave per 16-node tile; 3 chained WMMA, all operand loads unconditional b64.

__global__ void k_gemm1_wmma(const float* __restrict__ xp, const float* __restrict__ Wc1,
                             float* __restrict__ h1, int nTiles) {
    int wave = (int)((blockIdx.x * blockDim.x + threadIdx.x) >> 5);
    int lane = threadIdx.x & 31;
    if (wave >= nTiles) return;              // wave-uniform: EXEC stays all-1s
    int row0 = wave * 16;
    int m  = lane & 15;
    int hi = lane >> 4;                      // A/B layout: lanes 0-15 K pair {0,1}, lanes 16-31 {2,3}
    const float* xr = xp + (size_t)(row0 + m) * 12;
    const float* wr = Wc1 + m * 12;

    v8f c = {0.f, 0.f, 0.f, 0.f, 0.f, 0.f, 0.f, 0.f};
    #pragma unroll
    for (int kc = 0; kc < 3; ++kc) {
        int k0 = kc * 4 + hi * 2;
        v2f a = *(const v2f*)(xr + k0);
        v2f b = *(const v2f*)(wr + k0);
        c = __builtin_amdgcn_wmma_f32_16x16x4_f32(false, a, false, b, (short)0, c, false, false);
    }
    // C/D layout: VGPR j -> M = j (lanes 0-15) or j+8 (lanes 16-31); N = lane & 15
    #pragma unroll
    for (int j = 0; j < 8; ++j)
        h1[(size_t)(row0 + j + hi * 8) * 16 + m] = c[j];
}

// scalar tail for rows not covered by full 16-row tiles (empty when n % 16 == 0)
__global__ void k_gemm1_rem(const float* __restrict__ x, const float* __restrict__ W1,
                            float* __restrict__ h1, int rowStart, int n) {
    int idx = blockIdx.x * blockDim.x + threadIdx.x;   // over (n-rowStart)*16
    int r = rowStart + idx / 16, col = idx % 16;
    if (r >= n) return;
    float acc = 0.f;
    for (int k = 0; k < 10; ++k) acc += x[r * 10 + k] * W1[k * 16 + col];
    h1[r * 16 + col] = acc;
}

// ---------------- self-loop + bias init: agg1[i] = b1 + h1[i] * dinv[i]^2 ----------------

__global__ void k_init_agg1(const float* __restrict__ h1, const float* __restrict__ dinv,
                            const float* __restrict__ b1, float* __restrict__ agg1, int n) {
    int idx = blockIdx.x * blockDim.x + threadIdx.x;   // over n*16
    if (idx >= n * 16) return;
    int i = idx >> 4, cch = idx & 15;
    float w = dinv[i];
    agg1[idx] = b1[cch] + h1[idx] * w * w;
}

// ---------------- layer 1 scatter: one thread per (edge, component) ----------------
// 16 lanes form one edge group: h1[src] row read is one coalesced 64B line and the
// 16 atomics hit 1-2 cachelines of agg1[dst], so L2 can merge them.

__global__ void k_scatter1(const long long* __restrict__ src, const long long* __restrict__ dst,
                           const float* __restrict__ h1, const float* __restrict__ dinv,
                           float* __restrict__ agg1, int e) {
    int idx = blockIdx.x * blockDim.x + threadIdx.x;   // over e*16
    int ed = idx >> 4, c = idx & 15;
    if (ed >= e) return;
    int s = (int)src[ed], d = (int)dst[ed];
    float w = dinv[s] * dinv[d];
    atomAddF(agg1 + (size_t)d * 16 + c, h1[(size_t)s * 16 + c] * w);
}

// ---------------- layer 2 GEMM: h2 = relu(agg1) @ W2 via WMMA ----------------

__global__ void k_gemm2_wmma(const float* __restrict__ agg1, const float* __restrict__ Wc2,
                             float* __restrict__ h2, int nTiles) {
    int wave = (int)((blockIdx.x * blockDim.x + threadIdx.x) >> 5);
    int lane = threadIdx.x & 31;
    if (wave >= nTiles) return;
    int row0 = wave * 16;
    int m  = lane & 15;
    int hi = lane >> 4;
    const float* ar = agg1 + (size_t)(row0 + m) * 16;
    const float* wr = Wc2 + m * 16;

    v8f c = {0.f, 0.f, 0.f, 0.f, 0.f, 0.f, 0.f, 0.f};
    #pragma unroll
    for (int kc = 0; kc < 4; ++kc) {
        int k0 = kc * 4 + hi * 2;
        v2f a = *(const v2f*)(ar + k0);
        a.x = fmaxf(a.x, 0.f);                 // relu fused on A operand
        a.y = fmaxf(a.y, 0.f);
        v2f b = *(const v2f*)(wr + k0);
        c = __builtin_amdgcn_wmma_f32_16x16x4_f32(false, a, false, b, (short)0, c, false, false);
    }
    #pragma unroll
    for (int j = 0; j < 8; ++j)
        if (m < 2) h2[(size_t)(row0 + j + hi * 8) * 2 + m] = c[j];
}

__global__ void k_gemm2_rem(const float* __restrict__ agg1, const float* __restrict__ W2,
                            float* __restrict__ h2, int rowStart, int n) {
    int idx = blockIdx.x * blockDim.x + threadIdx.x;   // over (n-rowStart)*2
    int r = rowStart + idx / 2, col = idx % 2;
    if (r >= n) return;
    float acc = 0.f;
    for (int k = 0; k < 16; ++k) acc += fmaxf(agg1[r * 16 + k], 0.f) * W2[k * 2 + col];
    h2[r * 2 + col] = acc;
}

// ---------------- output init + layer 2 scatter ----------------

__global__ void k_init_out(const float* __restrict__ h2, const float* __restrict__ dinv,
                           const float* __restrict__ b2, float* __restrict__ out, int n) {
    int idx = blockIdx.x * blockDim.x + threadIdx.x;   // over n*2
    if (idx >= n * 2) return;
    int i = idx >> 1, cch = idx & 1;
    float w = dinv[i];
    out[idx] = b2[cch] + h2[idx] * w * w;
}

__global__ void k_scatter2(const long long* __restrict__ src, const long long* __restrict__ dst,
                           const float* __restrict__ h2, const float* __restrict__ dinv,
                           float* __restrict__ out, int e) {
    int i = blockIdx.x * blockDim.x + threadIdx.x;
    if (i >= e) return;
    int s = (int)src[i], d = (int)dst[i];
    float w = dinv[s] * dinv[d];
    float2 hv = *(const float2*)(h2 + (size_t)s * 2);
    atomAddF(out + (size_t)d * 2 + 0, hv.x * w);
    atomAddF(out + (size_t)d * 2 + 1, hv.y * w);
}

// ---------------- launch ----------------

extern "C" void kernel_launch(void* const* d_in, const int* in_sizes, int n_in,
                              void* d_out, int out_size, void* d_ws, size_t ws_size,
                              hipStream_t stream) {
    const float*     x   = (const float*)d_in[0];
    const long long* ei  = (const long long*)d_in[1];   // int64 [2, E] flat
    const float*     W1  = (const float*)d_in[2];
    const float*     b1  = (const float*)d_in[3];
    const float*     W2  = (const float*)d_in[4];
    const float*     b2  = (const float*)d_in[5];
    float*           out = (float*)d_out;

    const int n = in_sizes[0] / 10;
    const int e = in_sizes[1] / 2;
    const long long* src = ei;
    const long long* dst = ei + e;

    // workspace carve-up (all 256B aligned)
    char* ws = (char*)d_ws;
    size_t off = 0;
    auto carve = [&](size_t bytes) { char* p = ws + off; off = (off + bytes + 255) & ~(size_t)255; return p; };
    float* dinv = (float*)carve((size_t)n * 4);          //  0.8 MB (deg then rsqrt in-place)
    float* xp   = (float*)carve((size_t)n * 12 * 4);     //  9.6 MB padded x
    float* h1   = (float*)carve((size_t)n * 16 * 4);     // 12.8 MB
    float* agg1 = (float*)carve((size_t)n * 16 * 4);     // 12.8 MB
    float* h2   = (float*)carve((size_t)n * 2 * 4);      //  1.6 MB
    float* Wc1  = (float*)carve(192 * 4);                //  W1^T padded [16][12]
    float* Wc2  = (float*)carve(256 * 4);                //  W2^T padded [16][16]
    (void)ws_size;

    const int B = 256;
    const int fullTiles = n / 16;
    const int remRows   = n - fullTiles * 16;
    const int gTiles    = (fullTiles + (B / 32) - 1) / (B / 32);

    k_init_deg     <<<(n + B - 1) / B, B, 0, stream>>>(dinv, n);
    k_count_deg    <<<(e + B - 1) / B, B, 0, stream>>>(dst, dinv, e);
    k_finalize_dinv<<<(n + B - 1) / B, B, 0, stream>>>(dinv, n);

    k_pad_x        <<<(n * 12 + B - 1) / B, B, 0, stream>>>(x, xp, n);
    k_prep_w       <<<1, B, 0, stream>>>(W1, W2, Wc1, Wc2);

    if (fullTiles > 0)
        k_gemm1_wmma<<<gTiles, B, 0, stream>>>(xp, Wc1, h1, fullTiles);
    if (remRows > 0)
        k_gemm1_rem<<<(remRows * 16 + B - 1) / B, B, 0, stream>>>(x, W1, h1, fullTiles * 16, n);

    k_init_agg1    <<<(n * 16 + B - 1) / B, B, 0, stream>>>(h1, dinv, b1, agg1, n);
    k_scatter1     <<<(int)(((size_t)e * 16 + B - 1) / B), B, 0, stream>>>(src, dst, h1, dinv, agg1, e);

    if (fullTiles > 0)
        k_gemm2_wmma<<<gTiles, B, 0, stream>>>(agg1, Wc2, h2, fullTiles);
    if (remRows > 0)
        k_gemm2_rem<<<(remRows * 2 + B - 1) / B, B, 0, stream>>>(agg1, W2, h2, fullTiles * 16, n);

    k_init_out     <<<(n * 2 + B - 1) / B, B, 0, stream>>>(h2, dinv, b2, out, n);
    k_scatter2     <<<(e + B - 1) / B, B, 0, stream>>>(src, dst, h2, dinv, out, e);
}